// DiceLoss_38585986187842
// MI455X (gfx1250) — compile-verified
//
#include <hip/hip_runtime.h>

// DiceLoss (softmax + per-class dice) for x:(4,8,64,256,256) f32, target:(4,1,64,256,256) i64.
// Pass 1: fused softmax + per-(b,c) partial reductions, WMMA-based block reduction.
// Pass 2: deterministic cross-block reduction + dice formula -> scalar loss.

typedef float v2f __attribute__((ext_vector_type(2)));
typedef float v4f __attribute__((ext_vector_type(4)));
typedef float v8f __attribute__((ext_vector_type(8)));
typedef int   v4i __attribute__((ext_vector_type(4)));

#define NCLS         8
#define NBATCH       4
#define SPATIAL      (64 * 256 * 256)   // 4194304 voxels per (b,c)
#define BLOCKS_PER_B 512
#define THREADS      256
#define NACC         24                 // 8 psum + 8 inter + 8 count
#define LDS_STRIDE   32                 // padded row (cols 24..31 zeroed)

__global__ __launch_bounds__(THREADS)
void dice_partial_kernel(const float* __restrict__ x,
                         const int*   __restrict__ tgt32,   // int32 view of int64 labels
                         float*       __restrict__ partials) // [NBATCH][BLOCKS_PER_B][NACC]
{
    const int b   = blockIdx.y;
    const int blk = blockIdx.x;
    const int t   = threadIdx.x;

    const float* xb = x + (size_t)b * NCLS * SPATIAL;
    const v4i*   tb = reinterpret_cast<const v4i*>(tgt32) + (size_t)b * (SPATIAL / 2);

    float psum[NCLS]  = {0.f};
    float inter[NCLS] = {0.f};
    unsigned long long hist = 0ull;   // 8 packed 8-bit class counters (<=32 per thread)

    const float LOG2E  = 1.4426950408889634f;
    const int   stride = BLOCKS_PER_B * THREADS * 4;

    for (int s = (blk * THREADS + t) * 4; s < SPATIAL; s += stride) {
        // 8 channels x 4 voxels: 8x global_load_b128, nontemporal (streaming, no reuse)
        float xv[4][NCLS];
        #pragma unroll
        for (int c = 0; c < NCLS; ++c) {
            const v4f* xc = reinterpret_cast<const v4f*>(xb + (size_t)c * SPATIAL);
            v4f vv = __builtin_nontemporal_load(&xc[s >> 2]);
            xv[0][c] = vv.x; xv[1][c] = vv.y; xv[2][c] = vv.z; xv[3][c] = vv.w;
        }
        // 4 int64 labels: grab the low dwords (values in 0..7)
        v4i la = __builtin_nontemporal_load(&tb[(s >> 1)]);
        v4i lb = __builtin_nontemporal_load(&tb[(s >> 1) + 1]);
        int lab[4] = { la.x, la.z, lb.x, lb.z };

        #pragma unroll
        for (int j = 0; j < 4; ++j) {
            float m = xv[j][0];
            #pragma unroll
            for (int c = 1; c < NCLS; ++c) m = fmaxf(m, xv[j][c]);
            float mk = m * LOG2E;
            float e[NCLS];
            float denom = 0.f;
            #pragma unroll
            for (int c = 0; c < NCLS; ++c) {
                e[c] = __builtin_amdgcn_exp2f(__builtin_fmaf(xv[j][c], LOG2E, -mk));
                denom += e[c];
            }
            float inv = __builtin_amdgcn_rcpf(denom);
            int lj = lab[j];
            #pragma unroll
            for (int c = 0; c < NCLS; ++c) {
                float p = e[c] * inv;
                psum[c]  += p;
                inter[c] += (lj == c) ? p : 0.f;
            }
            hist += 1ull << (lj << 3);   // packed class histogram: 2 VALU ops vs 16
        }
    }

    // ---- block reduction: 256 rows x 24 cols -> 24 column sums via chained WMMA ----
    __shared__ float red[THREADS * LDS_STRIDE];
    {
        float* row = &red[t * LDS_STRIDE];
        #pragma unroll
        for (int c = 0; c < NCLS; ++c) {
            row[c]      = psum[c];
            row[8 + c]  = inter[c];
            row[16 + c] = (float)((hist >> (c << 3)) & 0xFFull);
            row[24 + c] = 0.f;     // keep padded cols clean of junk
        }
    }
    __syncthreads();

    if (t < 32) {   // wave 0 only: EXEC all-ones, as WMMA requires
        // A = ones(16x4): D[m][n] = sum_k B[k][n] (+C). All-ones A makes the
        // K-slot<->row mapping of B permutation-invariant; only lane->column
        // mapping matters (row striped across lanes within a VGPR).
        v2f aone = { 1.f, 1.f };
        v8f acc0 = { 0.f, 0.f, 0.f, 0.f, 0.f, 0.f, 0.f, 0.f };   // cols 0..15
        v8f acc1 = { 0.f, 0.f, 0.f, 0.f, 0.f, 0.f, 0.f, 0.f };   // cols 16..23 (+pad)
        const int lane = t;
        const int col  = lane & 15;
        const int rsel = (lane >> 4) << 1;   // lanes 0-15 -> rows r+0/r+1; 16-31 -> r+2/r+3
        for (int r = 0; r < THREADS; r += 4) {
            v2f b0, b1;
            b0.x = red[(r + rsel    ) * LDS_STRIDE + col];
            b0.y = red[(r + rsel + 1) * LDS_STRIDE + col];
            b1.x = red[(r + rsel    ) * LDS_STRIDE + 16 + col];
            b1.y = red[(r + rsel + 1) * LDS_STRIDE + 16 + col];
            acc0 = __builtin_amdgcn_wmma_f32_16x16x4_f32(false, aone, false, b0,
                                                         (short)0, acc0, false, false);
            acc1 = __builtin_amdgcn_wmma_f32_16x16x4_f32(false, aone, false, b1,
                                                         (short)0, acc1, false, false);
        }
        // D layout: element 0, lane n (n<16) = D[0][n] = column-sum n.
        float* outp = partials + ((size_t)b * BLOCKS_PER_B + blk) * NACC;
        if (lane < 16) outp[lane]      = acc0[0];
        if (lane < 8)  outp[16 + lane] = acc1[0];
    }
}

__global__ __launch_bounds__(128)
void dice_finalize_kernel(const float* __restrict__ partials,
                          float*       __restrict__ out)
{
    __shared__ float tot[NBATCH * NACC];
    const int t = threadIdx.x;
    if (t < NBATCH * NACC) {
        const int b = t / NACC, col = t % NACC;
        const float* p = partials + (size_t)b * BLOCKS_PER_B * NACC + col;
        float s = 0.f;
        for (int i = 0; i < BLOCKS_PER_B; ++i)   // fixed order -> deterministic
            s += p[(size_t)i * NACC];
        tot[t] = s;
    }
    __syncthreads();
    if (t == 0) {
        float acc = 0.f;
        for (int b = 0; b < NBATCH; ++b) {
            #pragma unroll
            for (int c = 0; c < NCLS; ++c) {
                float ps = tot[b * NACC + c];
                float in = tot[b * NACC + 8 + c];
                float cn = tot[b * NACC + 16 + c];
                acc += (2.f * in + 1.f) / (ps + cn + 1.f);
            }
        }
        out[0] = 1.f - acc / (float)(NBATCH * NCLS);
    }
}

extern "C" void kernel_launch(void* const* d_in, const int* in_sizes, int n_in,
                              void* d_out, int out_size, void* d_ws, size_t ws_size,
                              hipStream_t stream) {
    (void)in_sizes; (void)n_in; (void)out_size; (void)ws_size;
    const float* x     = (const float*)d_in[0];
    const int*   tgt32 = (const int*)d_in[1];       // int64 labels, int32 view
    float*       parts = (float*)d_ws;              // NBATCH*BLOCKS_PER_B*NACC floats (192 KB)
    float*       out   = (float*)d_out;

    dim3 grid(BLOCKS_PER_B, NBATCH, 1);
    dice_partial_kernel<<<grid, THREADS, 0, stream>>>(x, tgt32, parts);
    dice_finalize_kernel<<<1, 128, 0, stream>>>(parts, out);
}